// RelGraphSAGE_3332894621742
// MI455X (gfx1250) — compile-verified
//
#include <hip/hip_runtime.h>
#include <hip/hip_bf16.h>

typedef float v2f __attribute__((ext_vector_type(2)));
typedef float v8f __attribute__((ext_vector_type(8)));

#define IN_CH 128
#define HID   64

// out[n][h] = dot(x[n][:], W[h][:]) (+ bias[h] if has_bias)
// x: [nrows, 128] row-major, W: [64, 128] row-major, out: [nrows, 64]
// 256 threads = 8 wave32; each wave computes a 16x64 tile with
// V_WMMA_F32_16X16X4_F32 (full fp32, K-loop of 32 steps, 4 N-tiles).
__global__ void __launch_bounds__(256)
rgs_gemm_xWt(const float* __restrict__ x,
             const float* __restrict__ W,
             const float* __restrict__ bias,
             float* __restrict__ out,
             int nrows, int has_bias)
{
    __shared__ float sW[HID * IN_CH];  // 32 KB

    const int tid = threadIdx.x;
    #pragma unroll 4
    for (int i = tid; i < HID * IN_CH; i += 256) sW[i] = W[i];
    __syncthreads();

    const int wave = tid >> 5;
    const int lane = tid & 31;
    const int rowBase = (blockIdx.x * 8 + wave) * 16;
    if (rowBase >= nrows) return;      // wave-uniform: EXEC stays all-1s for WMMA

    const int m    = lane & 15;        // row / col within 16-tile
    const int koff = (lane >> 4) * 2;  // lanes 16-31 carry K+2,K+3 halves

    v8f acc0 = {}, acc1 = {}, acc2 = {}, acc3 = {};

    const float* __restrict__ xrow = x + (size_t)(rowBase + m) * IN_CH;

    for (int k0 = 0; k0 < IN_CH; k0 += 4) {
        v2f a;
        a.x = xrow[k0 + koff];
        a.y = xrow[k0 + koff + 1];

        v2f b0, b1, b2, b3;
        b0.x = sW[(0 * 16 + m) * IN_CH + k0 + koff];
        b0.y = sW[(0 * 16 + m) * IN_CH + k0 + koff + 1];
        b1.x = sW[(1 * 16 + m) * IN_CH + k0 + koff];
        b1.y = sW[(1 * 16 + m) * IN_CH + k0 + koff + 1];
        b2.x = sW[(2 * 16 + m) * IN_CH + k0 + koff];
        b2.y = sW[(2 * 16 + m) * IN_CH + k0 + koff + 1];
        b3.x = sW[(3 * 16 + m) * IN_CH + k0 + koff];
        b3.y = sW[(3 * 16 + m) * IN_CH + k0 + koff + 1];

        acc0 = __builtin_amdgcn_wmma_f32_16x16x4_f32(false, a, false, b0, (short)0, acc0, false, false);
        acc1 = __builtin_amdgcn_wmma_f32_16x16x4_f32(false, a, false, b1, (short)0, acc1, false, false);
        acc2 = __builtin_amdgcn_wmma_f32_16x16x4_f32(false, a, false, b2, (short)0, acc2, false, false);
        acc3 = __builtin_amdgcn_wmma_f32_16x16x4_f32(false, a, false, b3, (short)0, acc3, false, false);
    }

    float bv0 = 0.f, bv1 = 0.f, bv2 = 0.f, bv3 = 0.f;
    if (has_bias) {
        bv0 = bias[0 * 16 + m];
        bv1 = bias[1 * 16 + m];
        bv2 = bias[2 * 16 + m];
        bv3 = bias[3 * 16 + m];
    }

    // C/D layout: VGPR v -> row = v + 8*(lane>=16), col = lane&15
    const int rhalf = (lane >> 4) * 8;
    #pragma unroll
    for (int v = 0; v < 8; ++v) {
        const size_t row = (size_t)(rowBase + rhalf + v);
        float* o = out + row * HID;
        o[0 * 16 + m] = acc0[v] + bv0;
        o[1 * 16 + m] = acc1[v] + bv1;
        o[2 * 16 + m] = acc2[v] + bv2;
        o[3 * 16 + m] = acc3[v] + bv3;
    }
}

// One wave per edge; process only edges of relation r this pass.
// lane l handles h = l and h = l+32.
__global__ void __launch_bounds__(256)
rgs_scatter_rel(const int* __restrict__ erow,
                const int* __restrict__ ecol,
                const int* __restrict__ etype,
                const float* __restrict__ xp,     // [N, 64] projection for relation r
                const float* __restrict__ brel,   // b_rel + r*64
                float* __restrict__ agg,          // [N, 64]
                int nedges, int r)
{
    const int e = blockIdx.x * (blockDim.x >> 5) + (threadIdx.x >> 5);
    if (e >= nedges) return;
    if (etype[e] != r) return;

    const int lane = threadIdx.x & 31;
    const int src = erow[e];
    const int dst = ecol[e];

    const float m0 = xp[(size_t)src * HID + lane]      + brel[lane];
    const float m1 = xp[(size_t)src * HID + lane + 32] + brel[lane + 32];
    atomicAdd(&agg[(size_t)dst * HID + lane],      m0);
    atomicAdd(&agg[(size_t)dst * HID + lane + 32], m1);
}

// One wave per node: out[n] = dot(relu(agg[n]), W_out) + b_out
__global__ void __launch_bounds__(256)
rgs_final(const float* __restrict__ agg,
          const float* __restrict__ Wout,   // [64]
          const float* __restrict__ bout,   // [1]
          float* __restrict__ out, int nnodes)
{
    const int node = blockIdx.x * (blockDim.x >> 5) + (threadIdx.x >> 5);
    if (node >= nnodes) return;
    const int lane = threadIdx.x & 31;

    float v0 = agg[(size_t)node * HID + lane];
    float v1 = agg[(size_t)node * HID + lane + 32];
    v0 = v0 > 0.f ? v0 : 0.f;
    v1 = v1 > 0.f ? v1 : 0.f;
    float s = v0 * Wout[lane] + v1 * Wout[lane + 32];

    #pragma unroll
    for (int off = 16; off > 0; off >>= 1)
        s += __shfl_xor(s, off, 32);

    if (lane == 0) out[node] = s + bout[0];
}

extern "C" void kernel_launch(void* const* d_in, const int* in_sizes, int n_in,
                              void* d_out, int out_size, void* d_ws, size_t ws_size,
                              hipStream_t stream) {
    (void)n_in; (void)out_size; (void)ws_size;

    const float* x       = (const float*)d_in[0];   // [N, 128]
    const int*   eidx    = (const int*)  d_in[1];   // [2, E]
    const int*   etype   = (const int*)  d_in[2];   // [E]
    const float* W_rel   = (const float*)d_in[3];   // [8, 64, 128]
    const float* b_rel   = (const float*)d_in[4];   // [8, 64]
    const float* W_self  = (const float*)d_in[5];   // [64, 128]
    const float* b_self  = (const float*)d_in[6];   // [64]
    const float* W_out   = (const float*)d_in[7];   // [64]
    const float* b_out   = (const float*)d_in[8];   // [1]

    const int N = in_sizes[0] / IN_CH;
    const int E = in_sizes[2];
    const int R = in_sizes[4] / HID;                // 8 relations

    float* out = (float*)d_out;
    float* xp  = (float*)d_ws;                      // [N, 64] per-relation scratch
    float* agg = xp + (size_t)N * HID;              // [N, 64] accumulator

    const int* erow = eidx;
    const int* ecol = eidx + E;

    const int gemmGrid  = (N + 127) / 128;          // 8 wave-tiles of 16 rows / block
    const int edgeGrid  = (E + 7) / 8;              // 8 edges (waves) / block
    const int nodeGrid  = (N + 7) / 8;              // 8 nodes (waves) / block

    // Self projection + bias fully initializes agg (deterministic each call).
    rgs_gemm_xWt<<<gemmGrid, 256, 0, stream>>>(x, W_self, b_self, agg, N, 1);

    // Per relation: project all nodes, then scatter that relation's edges.
    for (int r = 0; r < R; ++r) {
        rgs_gemm_xWt<<<gemmGrid, 256, 0, stream>>>(
            x, W_rel + (size_t)r * HID * IN_CH, nullptr, xp, N, 0);
        rgs_scatter_rel<<<edgeGrid, 256, 0, stream>>>(
            erow, ecol, etype, xp, b_rel + (size_t)r * HID, agg, E, r);
    }

    rgs_final<<<nodeGrid, 256, 0, stream>>>(agg, W_out, b_out, out, N);
}